// MyModel_61933428412507
// MI455X (gfx1250) — compile-verified
//
#include <hip/hip_runtime.h>

// y = -0.1f * x, 64M fp32 elements. Pure streaming: HBM-bound at ~23 us on
// MI455X (512 MiB / 23.3 TB/s). No WMMA on purpose (AI = 0.125 flop/B).
// gfx1250 path exercised: global_load_async_to_lds_b128 (ASYNCcnt pipeline),
// s_wait_asynccnt, ds_load_b128, non-temporal b128 stores.

typedef float v4f __attribute__((ext_vector_type(4)));

#define TPB 256u                      // 8 wave32 per block
#define VPT 4u                        // float4 per thread per stage
#define STAGE_VEC (TPB * VPT)         // 1024 float4 = 16 KB per stage
#define MAX_BLOCKS 1024u

__device__ __forceinline__ void issue_stage_async(const void* src,
                                                  unsigned stageBaseVec,
                                                  unsigned n4,
                                                  unsigned ldsByteBase,
                                                  unsigned tid) {
  const unsigned long long sbase = (unsigned long long)src;
#pragma unroll
  for (unsigned j = 0; j < VPT; ++j) {
    unsigned v = stageBaseVec + tid + j * TPB;
    if (v > n4 - 1u) v = n4 - 1u;            // clamp: keep EXEC uniform around async op
    unsigned goff = v * 16u;                 // byte offset (<= 256 MiB, fits i32)
    unsigned lds  = ldsByteBase + (tid + j * TPB) * 16u;
    // GVS form: LDS-dest VGPR, 32-bit per-lane byte offset VGPR, 64-bit SGPR base.
    asm volatile("global_load_async_to_lds_b128 %0, %1, %2"
                 :
                 : "v"(lds), "v"(goff), "s"(sbase)
                 : "memory");
  }
}

__global__ __launch_bounds__(TPB) void scale_stream_async(const v4f* __restrict__ src,
                                                          v4f* __restrict__ dst,
                                                          unsigned n4, float scale) {
  __shared__ v4f sbuf[2][STAGE_VEC];         // 32 KB double buffer
  const unsigned tid = threadIdx.x;
  const unsigned nStages = (n4 + STAGE_VEC - 1u) / STAGE_VEC;
  unsigned s = blockIdx.x;
  if (s >= nStages) return;

  const unsigned ldsBase  = (unsigned)(size_t)(&sbuf[0][0]);  // low 32 bits = LDS offset
  const unsigned bufBytes = STAGE_VEC * 16u;

  // Prime the pipeline.
  issue_stage_async(src, s * STAGE_VEC, n4, ldsBase, tid);

  unsigned i = 0;
  for (; s < nStages; s += gridDim.x, ++i) {
    const unsigned snext = s + gridDim.x;
    if (snext < nStages) {
      issue_stage_async(src, snext * STAGE_VEC, n4,
                        ldsBase + ((i + 1u) & 1u) * bufBytes, tid);
      // 4 new loads outstanding; async loads complete in order, so <=4 means
      // the previous stage has fully landed in LDS.
      asm volatile("s_wait_asynccnt 0x4" ::: "memory");
    } else {
      asm volatile("s_wait_asynccnt 0x0" ::: "memory");
    }

    const v4f* cur = &sbuf[i & 1u][0];
    const unsigned base = s * STAGE_VEC;
#pragma unroll
    for (unsigned j = 0; j < VPT; ++j) {
      const unsigned v = base + tid + j * TPB;
      if (v < n4) {
        v4f d = cur[tid + j * TPB];          // ds_load_b128
        d *= scale;
        // NT store: output never re-read; keep L2 for input reuse across replays.
        __builtin_nontemporal_store(d, dst + v);
      }
    }
  }
}

__global__ void scale_tail(const float* __restrict__ src, float* __restrict__ dst,
                           unsigned start, unsigned n, float scale) {
  const unsigned idx = start + blockIdx.x * blockDim.x + threadIdx.x;
  if (idx < n) dst[idx] = src[idx] * scale;
}

extern "C" void kernel_launch(void* const* d_in, const int* in_sizes, int n_in,
                              void* d_out, int out_size, void* d_ws, size_t ws_size,
                              hipStream_t stream) {
  (void)n_in; (void)out_size; (void)d_ws; (void)ws_size;
  const float* x = (const float*)d_in[0];
  float* y = (float*)d_out;
  const unsigned n  = (unsigned)in_sizes[0];   // 67,108,864 for the reference shape
  const unsigned n4 = n / 4u;
  const float scale = -0.1f;

  if (n4) {
    const unsigned nStages = (n4 + STAGE_VEC - 1u) / STAGE_VEC;
    const unsigned grid = nStages < MAX_BLOCKS ? nStages : MAX_BLOCKS;
    scale_stream_async<<<grid, TPB, 0, stream>>>((const v4f*)x, (v4f*)y, n4, scale);
  }
  const unsigned rem = n % 4u;
  if (rem) {
    scale_tail<<<1, 64, 0, stream>>>(x, y, n4 * 4u, n, scale);
  }
}